// GNNCustomized_90099823935520
// MI455X (gfx1250) — compile-verified
//
#include <hip/hip_runtime.h>
#include <stdint.h>

typedef float v2f __attribute__((ext_vector_type(2)));
typedef float v8f __attribute__((ext_vector_type(8)));

#define DIM 128   // IN_DIM == HID_DIM == 128

// ---------------------------------------------------------------- degree
__global__ void k_deg_init(float* __restrict__ deg, int N) {
    int i = blockIdx.x * blockDim.x + threadIdx.x;
    if (i < N) deg[i] = 1.0f;   // self-loop contributes 1 at dst
}

__global__ void k_deg_edges(const long long* __restrict__ dst, float* __restrict__ deg, int E) {
    int e = blockIdx.x * blockDim.x + threadIdx.x;
    if (e < E) atomicAdd(&deg[(int)dst[e]], 1.0f);
}

__global__ void k_dinv(const float* __restrict__ deg, float* __restrict__ dinv, int N) {
    int i = blockIdx.x * blockDim.x + threadIdx.x;
    if (i < N) dinv[i] = rsqrtf(deg[i]);   // deg >= 1 always (self-loop)
}

// ---------------------------------------------------------------- layer-1 GEMM: h = x @ W1 (fp32 WMMA)
// Block = 256 threads = 8 waves. Wave w computes a 16-row strip of h across all
// 128 output columns (8 tiles of 16x16), accumulating over K=128 in steps of 4
// with V_WMMA_F32_16X16X4_F32.
//
// W1 is staged in LDS swizzled into K-pairs:
//   lw[(k>>1)*256 + col*2 + (k&1)] = W1[k][col]
// so each lane's B fragment {W1[ka][col], W1[ka+1][col]} is one aligned
// ds_load_b64 directly into the even register pair WMMA wants (no mov packing).
__global__ __launch_bounds__(256)
void k_gemm1(const float* __restrict__ x, const float* __restrict__ W1,
             float* __restrict__ h, int N, int nRowTiles) {
    __shared__ float lw[DIM * DIM];   // 64 KB, K-pair swizzled

    // cooperative load + swizzle of W1 into LDS (coalesced float4 global reads)
    {
        const float4* w4 = (const float4*)W1;
        for (int i = threadIdx.x; i < (DIM * DIM) / 4; i += 256) {
            float4 v = w4[i];
            int k  = i >> 5;          // 32 float4 per row of 128
            int c0 = (i & 31) * 4;    // first of 4 consecutive columns
            float* dst = lw + (k >> 1) * 256 + (k & 1);
            dst[(c0 + 0) * 2] = v.x;
            dst[(c0 + 1) * 2] = v.y;
            dst[(c0 + 2) * 2] = v.z;
            dst[(c0 + 3) * 2] = v.w;
        }
    }
    __syncthreads();

    const int wave = threadIdx.x >> 5;
    const int lane = threadIdx.x & 31;
    const int tile = blockIdx.x * 8 + wave;
    if (tile >= nRowTiles) return;          // wave-uniform: EXEC stays all-ones for WMMA

    const int row0  = tile * 16;
    const int lhalf = lane >> 4;            // 0: lanes 0-15, 1: lanes 16-31
    const int lmod  = lane & 15;

    // A fragment source row (clamp for generality; N % 16 == 0 here)
    int arow = row0 + lmod;
    if (arow >= N) arow = N - 1;
    const float* xrow = x + (size_t)arow * DIM;

    v8f acc[8];
#pragma unroll
    for (int n = 0; n < 8; ++n) acc[n] = (v8f){0.f, 0.f, 0.f, 0.f, 0.f, 0.f, 0.f, 0.f};

    for (int k = 0; k < DIM; k += 4) {
        // A 16x4 f32 layout: lanes 0-15 hold {K=k,k+1}, lanes 16-31 hold {K=k+2,k+3}
        const int ka = k + 2 * lhalf;                 // even
        v2f a = *(const v2f*)(xrow + ka);             // 8B-aligned global load

        const float* brow = lw + (ka >> 1) * 256;     // K-pair row in LDS
#pragma unroll
        for (int n = 0; n < 8; ++n) {
            // B 4x16 f32 layout mirrors C/D halving: lanes 0-15 rows {k,k+1},
            // lanes 16-31 rows {k+2,k+3}; pair is contiguous in LDS -> one b64 load
            const int col = n * 16 + lmod;
            v2f b = *(const v2f*)(brow + col * 2);
            acc[n] = __builtin_amdgcn_wmma_f32_16x16x4_f32(
                /*neg_a=*/false, a, /*neg_b=*/false, b,
                /*c_mod=*/(short)0, acc[n], /*reuse_a=*/false, /*reuse_b=*/false);
        }
    }

    // C/D 16x16 f32 layout: VGPR r -> row (r + 8*lhalf), col = lane%16
#pragma unroll
    for (int n = 0; n < 8; ++n) {
        const int col = n * 16 + lmod;
#pragma unroll
        for (int r = 0; r < 8; ++r) {
            const int row = row0 + r + 8 * lhalf;
            if (row < N) h[(size_t)row * DIM + col] = acc[n][r];
        }
    }
}

// ---------------------------------------------------------------- aggregation layer 1
__global__ void k_zero_a1(float* __restrict__ a1, int total) {
    int i = blockIdx.x * blockDim.x + threadIdx.x;
    if (i < total) a1[i] = 0.0f;
}

// One wave per edge: src/dst loads are wave-uniform (scalarized), each lane
// gathers a float4 (512 B coalesced per edge) and issues 4 global float atomics.
__global__ void k_scatter1(const float* __restrict__ h, const float* __restrict__ dinv,
                           const long long* __restrict__ src, const long long* __restrict__ dst,
                           float* __restrict__ a1, int E) {
    int t = blockIdx.x * blockDim.x + threadIdx.x;
    int e = t >> 5;
    if (e >= E) return;
    int part = t & 31;
    int s = (int)src[e];
    int d = (int)dst[e];
    float nrm = dinv[s] * dinv[d];
    float4 hv = *(const float4*)(h + (size_t)s * DIM + part * 4);
    float* base = a1 + (size_t)d * DIM + part * 4;
    atomicAdd(base + 0, hv.x * nrm);
    atomicAdd(base + 1, hv.y * nrm);
    atomicAdd(base + 2, hv.z * nrm);
    atomicAdd(base + 3, hv.w * nrm);
}

// Fold self-loop contribution (h[i]*dinv[i]^2) + bias + ReLU; avoids self-loop atomics.
__global__ void k_relu1(float* __restrict__ a1, const float* __restrict__ h,
                        const float* __restrict__ dinv, const float* __restrict__ b1, int total) {
    int i = blockIdx.x * blockDim.x + threadIdx.x;
    if (i >= total) return;
    int node = i >> 7;
    int dch  = i & (DIM - 1);
    float di = dinv[node];
    float v  = a1[i] + h[i] * di * di + b1[dch];
    a1[i] = v > 0.0f ? v : 0.0f;
}

// ---------------------------------------------------------------- layer-2 mat-vec: h2 = a1 @ W2
// Wave per node: 32 lanes x float4 dot, butterfly reduce (wave32).
__global__ __launch_bounds__(256)
void k_gemm2(const float* __restrict__ a1, const float* __restrict__ W2,
             float* __restrict__ h2, int N) {
    int wave = threadIdx.x >> 5;
    int lane = threadIdx.x & 31;
    int node = blockIdx.x * 8 + wave;
    if (node >= N) return;
    float4 a = ((const float4*)(a1 + (size_t)node * DIM))[lane];
    float4 w = ((const float4*)W2)[lane];
    float s = a.x * w.x + a.y * w.y + a.z * w.z + a.w * w.w;
#pragma unroll
    for (int off = 16; off > 0; off >>= 1) s += __shfl_xor(s, off, 32);
    if (lane == 0) h2[node] = s;
}

__global__ void k_out_init(const float* __restrict__ h2, const float* __restrict__ dinv,
                           const float* __restrict__ b2, float* __restrict__ out, int N) {
    int i = blockIdx.x * blockDim.x + threadIdx.x;
    if (i >= N) return;
    float di = dinv[i];
    out[i] = h2[i] * di * di + b2[0];   // self-loop term + bias
}

__global__ void k_scatter2(const float* __restrict__ h2, const float* __restrict__ dinv,
                           const long long* __restrict__ src, const long long* __restrict__ dst,
                           float* __restrict__ out, int E) {
    int e = blockIdx.x * blockDim.x + threadIdx.x;
    if (e >= E) return;
    int s = (int)src[e];
    int d = (int)dst[e];
    atomicAdd(&out[d], h2[s] * dinv[s] * dinv[d]);
}

// ---------------------------------------------------------------- launcher
extern "C" void kernel_launch(void* const* d_in, const int* in_sizes, int n_in,
                              void* d_out, int out_size, void* d_ws, size_t ws_size,
                              hipStream_t stream) {
    const float*     x   = (const float*)d_in[0];
    const long long* ei  = (const long long*)d_in[1];   // int64 edge_index [2, E]
    const float*     W1  = (const float*)d_in[2];
    const float*     b1  = (const float*)d_in[3];
    const float*     W2  = (const float*)d_in[4];
    const float*     b2  = (const float*)d_in[5];
    float*           out = (float*)d_out;

    const int N = in_sizes[0] / DIM;       // 100000
    const int E = in_sizes[1] / 2;         // 1600000
    const long long* srce = ei;
    const long long* dste = ei + E;

    // workspace layout (floats): deg[N] | dinv[N] | h[N*128] | a1[N*128] | h2[N]
    float* ws   = (float*)d_ws;
    float* deg  = ws;
    float* dinv = deg + N;
    float* h    = dinv + N;
    float* a1   = h + (size_t)N * DIM;
    float* h2   = a1 + (size_t)N * DIM;

    const int B = 256;
    const int totalND = N * DIM;
    const int nRowTiles = (N + 15) / 16;

    // norm
    k_deg_init <<<(N + B - 1) / B, B, 0, stream>>>(deg, N);
    k_deg_edges<<<(E + B - 1) / B, B, 0, stream>>>(dste, deg, E);
    k_dinv     <<<(N + B - 1) / B, B, 0, stream>>>(deg, dinv, N);

    // layer 1
    k_gemm1    <<<(nRowTiles + 7) / 8, B, 0, stream>>>(x, W1, h, N, nRowTiles);
    k_zero_a1  <<<(totalND + B - 1) / B, B, 0, stream>>>(a1, totalND);
    {
        long long threads = (long long)E * 32;
        k_scatter1<<<(unsigned)((threads + B - 1) / B), B, 0, stream>>>(h, dinv, srce, dste, a1, E);
    }
    k_relu1    <<<(totalND + B - 1) / B, B, 0, stream>>>(a1, h, dinv, b1, totalND);

    // layer 2
    k_gemm2    <<<(N + 7) / 8, B, 0, stream>>>(a1, W2, h2, N);
    k_out_init <<<(N + B - 1) / B, B, 0, stream>>>(h2, dinv, b2, out, N);
    k_scatter2 <<<(E + B - 1) / B, B, 0, stream>>>(h2, dinv, srce, dste, out, E);
}